// TimeIntervalAwareTransformer_66099546685526
// MI455X (gfx1250) — compile-verified
//
#include <hip/hip_runtime.h>
#include <hip/hip_bf16.h>
#include <math.h>

// Problem constants (from setup_inputs)
#define BB 4
#define SS 160
#define DD 256
#define UU 128
#define PP 10000
#define NEGV (-4294967295.0f)
#define LDAW 164   // padded LDS row stride for attention scores

typedef __attribute__((ext_vector_type(2))) float v2f;
typedef __attribute__((ext_vector_type(8))) float v8f;

__device__ __forceinline__ int iabs(int x) { return x < 0 ? -x : x; }

// fp32 WMMA: D = A(16x4) * B(4x16) + C(16x16)
__device__ __forceinline__ v8f wmma4(v2f a, v2f b, v8f c) {
    return __builtin_amdgcn_wmma_f32_16x16x4_f32(
        false, a, false, b, (short)0, c, false, false);
}

// -------------------------------------------------------------------------
// Generic GEMM: C[M,N] = A[M,K] @ W[N,K]^T + bias[N]  (optionally ReLU)
// One wave per 16x16 tile; 4 waves per block cover 4 consecutive N-tiles.
// Rows are CLAMPED (not zeroed): OOB rows only affect C elements that are
// never stored, so the inner loop is fully branch-free (EXEC stays all-1s).
// -------------------------------------------------------------------------
__global__ void gemm_kernel(const float* __restrict__ A,
                            const float* __restrict__ W,
                            const float* __restrict__ bias,
                            float* __restrict__ C,
                            int M, int N, int K, int relu)
{
    int lane = threadIdx.x & 31;
    int wave = threadIdx.x >> 5;
    int nt = blockIdx.x * 4 + wave;
    int mt = blockIdx.y;
    int row0 = mt * 16, col0 = nt * 16;
    int l16 = lane & 15;
    int hi  = lane >> 4;
    int koff = hi << 1;                 // 0 or 2
    int m = row0 + l16;
    int n = col0 + l16;
    bool nv = (n < N);
    int mc = m < M ? m : (M - 1);       // clamped rows: branch-free loads
    int nc = n < N ? n : (N - 1);

    const float* Arow = A + (size_t)mc * K + koff;
    const float* Wrow = W + (size_t)nc * K + koff;

    v8f c = {};
    // software pipeline: preload next fragments before issuing current WMMA
    v2f a = *(const v2f*)(Arow);
    v2f b = *(const v2f*)(Wrow);
    #pragma unroll 4
    for (int k0 = 0; k0 < K - 4; k0 += 4) {
        v2f an = *(const v2f*)(Arow + k0 + 4);
        v2f bn = *(const v2f*)(Wrow + k0 + 4);
        c = wmma4(a, b, c);
        a = an; b = bn;
    }
    c = wmma4(a, b, c);

    float bn_ = (nv && bias) ? bias[n] : 0.f;
    for (int r = 0; r < 8; ++r) {
        int mm = row0 + r + (hi << 3);
        if (mm < M && nv) {
            float v = c[r] + bn_;
            if (relu) v = fmaxf(v, 0.f);
            C[(size_t)mm * N + n] = v;
        }
    }
}

// -------------------------------------------------------------------------
// N-blocked GEMM for the wide decoder GEMM: each wave computes a 16x64
// tile (4 N-tiles), amortizing the A-fragment load across 4 WMMAs
// (load:wmma ratio 1.25 instead of 2). Branch-free inner loop (clamped rows).
// Block = 4 waves -> 256 columns per block.
// -------------------------------------------------------------------------
__global__ void gemm4_kernel(const float* __restrict__ A,
                             const float* __restrict__ W,
                             const float* __restrict__ bias,
                             float* __restrict__ C,
                             int M, int N, int K)
{
    int lane = threadIdx.x & 31;
    int wave = threadIdx.x >> 5;
    int nt0 = (blockIdx.x * 4 + wave) * 4;     // first of 4 N-tiles
    int mt  = blockIdx.y;
    int row0 = mt * 16;
    int l16 = lane & 15;
    int hi  = lane >> 4;
    int koff = hi << 1;
    int m = row0 + l16;
    int mc = m < M ? m : (M - 1);
    const float* Arow = A + (size_t)mc * K + koff;

    const float* Wrow[4];
    int ncol[4];
    #pragma unroll
    for (int t = 0; t < 4; ++t) {
        int n = (nt0 + t) * 16 + l16;
        ncol[t] = n;
        int nc = n < N ? n : (N - 1);
        Wrow[t] = W + (size_t)nc * K + koff;
    }

    v8f c0 = {}, c1 = {}, c2 = {}, c3 = {};
    v2f a = *(const v2f*)(Arow);
    v2f b0 = *(const v2f*)(Wrow[0]);
    v2f b1 = *(const v2f*)(Wrow[1]);
    v2f b2 = *(const v2f*)(Wrow[2]);
    v2f b3 = *(const v2f*)(Wrow[3]);
    #pragma unroll 2
    for (int k0 = 0; k0 < K - 4; k0 += 4) {
        v2f an  = *(const v2f*)(Arow + k0 + 4);
        v2f bn0 = *(const v2f*)(Wrow[0] + k0 + 4);
        v2f bn1 = *(const v2f*)(Wrow[1] + k0 + 4);
        v2f bn2 = *(const v2f*)(Wrow[2] + k0 + 4);
        v2f bn3 = *(const v2f*)(Wrow[3] + k0 + 4);
        c0 = wmma4(a, b0, c0);
        c1 = wmma4(a, b1, c1);
        c2 = wmma4(a, b2, c2);
        c3 = wmma4(a, b3, c3);
        a = an; b0 = bn0; b1 = bn1; b2 = bn2; b3 = bn3;
    }
    c0 = wmma4(a, b0, c0);
    c1 = wmma4(a, b1, c1);
    c2 = wmma4(a, b2, c2);
    c3 = wmma4(a, b3, c3);

    v8f acc[4] = {c0, c1, c2, c3};
    #pragma unroll
    for (int t = 0; t < 4; ++t) {
        int n = ncol[t];
        if (n >= N) continue;
        float bn_ = bias ? bias[n] : 0.f;
        #pragma unroll
        for (int r = 0; r < 8; ++r) {
            int mm = row0 + r + (hi << 3);
            if (mm < M)
                C[(size_t)mm * N + n] = acc[t][r] + bn_;
        }
    }
}

// -------------------------------------------------------------------------
// RoPE: out[row, :] = rope(in[row, :]) with pos = row % S
// -------------------------------------------------------------------------
__global__ void rope_kernel(const float* __restrict__ in, float* __restrict__ out)
{
    int row = blockIdx.x;                 // b*S + pos
    int pos = row % SS;
    int p = threadIdx.x;                  // 0..127 (D/2 pairs)
    const float* r = in + (size_t)row * DD;
    float t0 = r[2 * p], t1 = r[2 * p + 1];
    float inv = __powf(10000.0f, -(float)(2 * p) / (float)DD);
    float f = (float)pos * inv;
    float cs = __cosf(f), sn = __sinf(f);
    out[(size_t)row * DD + 2 * p]     = t0 * cs - t1 * sn;
    out[(size_t)row * DD + 2 * p + 1] = t1 * cs + t0 * sn;
}

// -------------------------------------------------------------------------
// Attention block: per (b, i-tile) workgroup (256 thr = 8 waves).
// scores (WMMA) -> +table bias -> scale -> mask -> softmax -> x = aw@V (WMMA)
// + wsum@hourE + wsum@dayE  (table-collapsed einsums)
// -------------------------------------------------------------------------
__global__ void attn_kernel(const float* __restrict__ Q,
                            const float* __restrict__ Km,
                            const float* __restrict__ V,
                            const float* __restrict__ qh,   // [B*S, 26]
                            const float* __restrict__ qd,   // [B*S, 8]
                            const float* __restrict__ hourE,
                            const float* __restrict__ dayE,
                            const int* __restrict__ in_h,
                            const int* __restrict__ in_w,
                            const int* __restrict__ seq_lens,
                            float* __restrict__ Xout)
{
    __shared__ float aw[16 * LDAW];
    __shared__ float wsh[16 * 26];
    __shared__ float wsd[16 * 8];

    int blk = blockIdx.x;
    int b  = blk / (SS / 16);
    int it = blk % (SS / 16);
    int i0 = it * 16;
    int tid  = threadIdx.x;
    int lane = tid & 31;
    int wave = tid >> 5;
    int l16 = lane & 15;
    int hi  = lane >> 4;
    int koff = hi << 1;
    int slen = seq_lens[b];

    // 1) scores: aw[i,j] = Q[i,:] . K[j,:]   (10 j-tiles over 8 waves)
    const float* Arow = Q + ((size_t)b * SS + i0 + l16) * DD + koff;
    for (int jt = wave; jt < SS / 16; jt += 8) {
        int j0 = jt * 16;
        const float* Krow = Km + ((size_t)b * SS + j0 + l16) * DD + koff;
        v8f c = {};
        v2f a = *(const v2f*)(Arow);
        v2f bb = *(const v2f*)(Krow);
        #pragma unroll 4
        for (int k0 = 0; k0 < DD - 4; k0 += 4) {
            v2f an = *(const v2f*)(Arow + k0 + 4);
            v2f bn = *(const v2f*)(Krow + k0 + 4);
            c = wmma4(a, bb, c);
            a = an; bb = bn;
        }
        c = wmma4(a, bb, c);
        for (int r = 0; r < 8; ++r) {
            int i = r + (hi << 3);
            aw[i * LDAW + j0 + l16] = c[r];
        }
    }
    for (int t = tid; t < 16 * 26; t += 256) wsh[t] = 0.f;
    for (int t = tid; t < 16 * 8;  t += 256) wsd[t] = 0.f;
    __syncthreads();

    // 2) add table bias, scale, mask
    for (int t = tid; t < 16 * SS; t += 256) {
        int i = t / SS, j = t % SS;
        int gi = i0 + i;
        float v = aw[i * LDAW + j];
        bool valid = (j <= gi) && (gi < slen);
        int hI = 0, dI = 0;
        if (valid) {
            hI = (b == gi) ? 1 : (iabs(in_h[b * SS + gi] - in_h[b * SS + j]) + 2);
            dI = iabs(in_w[b * SS + gi] - in_w[b * SS + j]) + 1;
        }
        v += qh[((size_t)b * SS + gi) * 26 + hI] + qd[((size_t)b * SS + gi) * 8 + dI];
        v *= (1.0f / 16.0f);               // 1/sqrt(256)
        bool masked = (gi >= slen) || (j > gi);
        aw[i * LDAW + j] = masked ? NEGV : v;
    }
    __syncthreads();

    // 3) softmax per row (2 rows per wave) + table-weight accumulation
    for (int i = wave; i < 16; i += 8) {
        int gi = i0 + i;
        float mx = -INFINITY;
        for (int j = lane; j < SS; j += 32) mx = fmaxf(mx, aw[i * LDAW + j]);
        for (int o = 16; o > 0; o >>= 1) mx = fmaxf(mx, __shfl_xor(mx, o, 32));
        float sum = 0.f;
        for (int j = lane; j < SS; j += 32) {
            float e = __expf(aw[i * LDAW + j] - mx);
            aw[i * LDAW + j] = e;
            sum += e;
        }
        for (int o = 16; o > 0; o >>= 1) sum += __shfl_xor(sum, o, 32);
        float inv = 1.0f / sum;
        for (int j = lane; j < SS; j += 32) {
            float w = aw[i * LDAW + j] * inv;
            aw[i * LDAW + j] = w;
            bool valid = (j <= gi) && (gi < slen);
            int hI = 0, dI = 0;
            if (valid) {
                hI = (b == gi) ? 1 : (iabs(in_h[b * SS + gi] - in_h[b * SS + j]) + 2);
                dI = iabs(in_w[b * SS + gi] - in_w[b * SS + j]) + 1;
            }
            atomicAdd(&wsh[i * 26 + hI], w);
            atomicAdd(&wsd[i * 8 + dI], w);
        }
    }
    __syncthreads();

    // 4) x = aw @ V  (+ wsum @ hourE + wsum @ dayE); 16 d-tiles over 8 waves
    for (int dt = wave; dt < DD / 16; dt += 8) {
        int d0 = dt * 16;
        int n = d0 + l16;
        v8f c = {};
        #pragma unroll 4
        for (int k0 = 0; k0 < SS; k0 += 4) {
            int ka = k0 + koff;
            v2f a; a.x = aw[l16 * LDAW + ka]; a.y = aw[l16 * LDAW + ka + 1];
            v2f bb;
            bb.x = V[((size_t)b * SS + ka)     * DD + n];
            bb.y = V[((size_t)b * SS + ka + 1) * DD + n];
            c = wmma4(a, bb, c);
        }
        for (int r = 0; r < 8; ++r) {
            int i = r + (hi << 3);
            float v = c[r];
            for (int e = 0; e < 26; ++e) v += wsh[i * 26 + e] * hourE[e * DD + n];
            for (int e = 0; e < 8;  ++e) v += wsd[i * 8 + e]  * dayE[e * DD + n];
            Xout[((size_t)b * SS + i0 + i) * DD + n] = v;
        }
    }
}

// -------------------------------------------------------------------------
// out = LayerNorm(x + skip), no affine, eps=1e-5. One block per row (256 thr)
// -------------------------------------------------------------------------
__global__ void add_ln_kernel(const float* __restrict__ x,
                              const float* __restrict__ skip,
                              float* __restrict__ out)
{
    int row = blockIdx.x;
    int d = threadIdx.x;
    int lane = d & 31, wave = d >> 5;
    __shared__ float red[8];
    float v = x[(size_t)row * DD + d] + skip[(size_t)row * DD + d];
    float s = v;
    for (int o = 16; o > 0; o >>= 1) s += __shfl_xor(s, o, 32);
    if (lane == 0) red[wave] = s;
    __syncthreads();
    float mean = 0.f;
    for (int k = 0; k < 8; ++k) mean += red[k];
    mean *= (1.0f / DD);
    float dv = v - mean;
    float q = dv * dv;
    __syncthreads();
    for (int o = 16; o > 0; o >>= 1) q += __shfl_xor(q, o, 32);
    if (lane == 0) red[wave] = q;
    __syncthreads();
    float var = 0.f;
    for (int k = 0; k < 8; ++k) var += red[k];
    var *= (1.0f / DD);
    out[(size_t)row * DD + d] = dv * rsqrtf(var + 1e-5f);
}

// -------------------------------------------------------------------------
// base[b,j] = X[b,j,:] . u[b,:]   (one wave per row, 8 waves/block)
// -------------------------------------------------------------------------
__global__ void rowdot_kernel(const float* __restrict__ X,
                              const float* __restrict__ u,
                              float* __restrict__ out)
{
    int row = blockIdx.x * 8 + (threadIdx.x >> 5);
    int lane = threadIdx.x & 31;
    if (row >= BB * SS) return;
    int b = row / SS;
    float s = 0.f;
    for (int k = lane; k < DD; k += 32)
        s += X[(size_t)row * DD + k] * u[(size_t)b * DD + k];
    for (int o = 16; o > 0; o >>= 1) s += __shfl_xor(s, o, 32);
    if (lane == 0) out[row] = s;
}

// -------------------------------------------------------------------------
// User attention + aggregation. One block per (b,i). 256 threads.
// sc[j] = base[b,j] + lhu[b,lhI] + ldu[b,ldI] (j<=i else NEG); softmax;
// agg = sum_j w*ropex[b,j,:] + wsum_lh@lhourE + wsum_ld@ldayE
// -------------------------------------------------------------------------
__global__ void useragg_kernel(const float* __restrict__ ropex,
                               const float* __restrict__ base,
                               const float* __restrict__ lhu,  // [B,26]
                               const float* __restrict__ ldu,  // [B,8]
                               const float* __restrict__ lhourE,
                               const float* __restrict__ ldayE,
                               const int* __restrict__ in_h,
                               const int* __restrict__ in_w,
                               const int* __restrict__ lab_h,
                               const int* __restrict__ lab_w,
                               const int* __restrict__ seq_lens,
                               float* __restrict__ agg)
{
    int row = blockIdx.x;      // b*S + i
    int b = row / SS, i = row % SS;
    int tid = threadIdx.x;
    int lane = tid & 31, wave = tid >> 5;
    __shared__ float w[SS];
    __shared__ float wlh[26];
    __shared__ float wld[8];
    __shared__ float red[8];
    int slen = seq_lens[b];

    if (tid < SS) {
        int j = tid;
        float sc;
        if (j <= i) {
            bool valid = (i < slen);   // (j<=i) already true here
            int lhI = valid ? (iabs(lab_h[b * SS + i] - in_h[b * SS + j]) + 2) : 0;
            int ldI = valid ? (iabs(lab_w[b * SS + i] - in_w[b * SS + j]) + 1) : 0;
            sc = base[b * SS + j] + lhu[b * 26 + lhI] + ldu[b * 8 + ldI];
        } else sc = NEGV;
        w[j] = sc;
    }
    if (tid < 26) wlh[tid] = 0.f;
    if (tid < 8)  wld[tid] = 0.f;
    __syncthreads();

    // softmax over 160 entries
    float val = (tid < SS) ? w[tid] : -INFINITY;
    for (int o = 16; o > 0; o >>= 1) val = fmaxf(val, __shfl_xor(val, o, 32));
    if (lane == 0) red[wave] = val;
    __syncthreads();
    float mx = -INFINITY;
    for (int k = 0; k < 8; ++k) mx = fmaxf(mx, red[k]);
    __syncthreads();
    float e = (tid < SS) ? __expf(w[tid] - mx) : 0.f;
    float ssum = e;
    for (int o = 16; o > 0; o >>= 1) ssum += __shfl_xor(ssum, o, 32);
    if (lane == 0) red[wave] = ssum;
    __syncthreads();
    float tot = 0.f;
    for (int k = 0; k < 8; ++k) tot += red[k];
    float inv = 1.0f / tot;
    if (tid < SS) {
        int j = tid;
        float ww = e * inv;
        w[j] = ww;
        bool valid = (j <= i) && (i < slen);
        int lhI = valid ? (iabs(lab_h[b * SS + i] - in_h[b * SS + j]) + 2) : 0;
        int ldI = valid ? (iabs(lab_w[b * SS + i] - in_w[b * SS + j]) + 1) : 0;
        atomicAdd(&wlh[lhI], ww);
        atomicAdd(&wld[ldI], ww);
    }
    __syncthreads();

    int d = tid;  // 0..255
    float acc = 0.f;
    for (int j = 0; j < SS; ++j)
        acc += w[j] * ropex[((size_t)b * SS + j) * DD + d];
    for (int e2 = 0; e2 < 26; ++e2) acc += wlh[e2] * lhourE[e2 * DD + d];
    for (int e2 = 0; e2 < 8;  ++e2) acc += wld[e2] * ldayE[e2 * DD + d];
    agg[(size_t)row * DD + d] = acc;
}

// -------------------------------------------------------------------------
static void launch_gemm(const float* A, const float* W, const float* bias,
                        float* C, int M, int N, int K, int relu, hipStream_t s)
{
    dim3 grid((N + 63) / 64, (M + 15) / 16);
    gemm_kernel<<<grid, 128, 0, s>>>(A, W, bias, C, M, N, K, relu);
}

static void launch_gemm4(const float* A, const float* W, const float* bias,
                         float* C, int M, int N, int K, hipStream_t s)
{
    dim3 grid((N + 255) / 256, (M + 15) / 16);
    gemm4_kernel<<<grid, 128, 0, s>>>(A, W, bias, C, M, N, K);
}

extern "C" void kernel_launch(void* const* d_in, const int* in_sizes, int n_in,
                              void* d_out, int out_size, void* d_ws, size_t ws_size,
                              hipStream_t stream)
{
    (void)in_sizes; (void)n_in; (void)out_size; (void)ws_size;
    const float* src      = (const float*)d_in[0];
    const float* user_emb = (const float*)d_in[1];
    const int*   seq_lens = (const int*)d_in[2];
    const int*   in_h     = (const int*)d_in[3];
    const int*   in_w     = (const int*)d_in[4];
    const int*   lab_h    = (const int*)d_in[5];
    const int*   lab_w    = (const int*)d_in[6];
    const float* W1q = (const float*)d_in[7];  const float* b1q = (const float*)d_in[8];
    const float* W1k = (const float*)d_in[9];  const float* b1k = (const float*)d_in[10];
    const float* W1v = (const float*)d_in[11]; const float* b1v = (const float*)d_in[12];
    const float* F1a = (const float*)d_in[13]; const float* F1ab = (const float*)d_in[14];
    const float* F1b = (const float*)d_in[15]; const float* F1bb = (const float*)d_in[16];
    const float* W2q = (const float*)d_in[17]; const float* b2q = (const float*)d_in[18];
    const float* W2k = (const float*)d_in[19]; const float* b2k = (const float*)d_in[20];
    const float* W2v = (const float*)d_in[21]; const float* b2v = (const float*)d_in[22];
    const float* F2a = (const float*)d_in[23]; const float* F2ab = (const float*)d_in[24];
    const float* F2b = (const float*)d_in[25]; const float* F2bb = (const float*)d_in[26];
    const float* upW = (const float*)d_in[27]; const float* upb = (const float*)d_in[28];
    const float* decW = (const float*)d_in[29]; const float* decb = (const float*)d_in[30];
    const float* hourE  = (const float*)d_in[31];
    const float* dayE   = (const float*)d_in[32];
    const float* lhourE = (const float*)d_in[33];
    const float* ldayE  = (const float*)d_in[34];
    float* out = (float*)d_out;

    const size_t R = (size_t)BB * SS * DD;   // 163840 floats
    float* ws = (float*)d_ws;
    float* s0   = ws + 0 * R;
    float* Qb   = ws + 1 * R;
    float* Kb   = ws + 2 * R;
    float* Vb   = ws + 3 * R;
    float* xa   = ws + 4 * R;
    float* x1   = ws + 5 * R;
    float* hf   = ws + 6 * R;
    float* yb   = ws + 7 * R;
    float* x2   = ws + 8 * R;
    float* rx2  = ws + 9 * R;
    float* aggb = ws + 10 * R;
    float* qh   = ws + 11 * R;                 // 640*26
    float* qd   = qh + (size_t)BB * SS * 26;   // 640*8
    float* u    = qd + (size_t)BB * SS * 8;    // 4*256
    float* lhu  = u + BB * DD;                 // 4*26
    float* ldu  = lhu + BB * 26;               // 4*8
    float* base = ldu + BB * 8;                // 640

    const int MR = BB * SS;   // 640

    rope_kernel<<<MR, 128, 0, stream>>>(src, s0);

    // ---- layer 1 ----
    launch_gemm(s0, W1q, b1q, Qb, MR, DD, DD, 0, stream);
    launch_gemm(s0, W1k, b1k, Kb, MR, DD, DD, 0, stream);
    launch_gemm(s0, W1v, b1v, Vb, MR, DD, DD, 0, stream);
    launch_gemm(Qb, hourE, nullptr, qh, MR, 26, DD, 0, stream);
    launch_gemm(Qb, dayE,  nullptr, qd, MR, 8,  DD, 0, stream);
    attn_kernel<<<BB * (SS / 16), 256, 0, stream>>>(Qb, Kb, Vb, qh, qd, hourE, dayE,
                                                    in_h, in_w, seq_lens, xa);
    add_ln_kernel<<<MR, 256, 0, stream>>>(xa, s0, x1);
    launch_gemm(x1, F1a, F1ab, hf, MR, DD, DD, 1, stream);
    launch_gemm(hf, F1b, F1bb, yb, MR, DD, DD, 0, stream);
    add_ln_kernel<<<MR, 256, 0, stream>>>(yb, x1, x2);

    // ---- layer 2 ----
    launch_gemm(x2, W2q, b2q, Qb, MR, DD, DD, 0, stream);
    launch_gemm(x2, W2k, b2k, Kb, MR, DD, DD, 0, stream);
    launch_gemm(x2, W2v, b2v, Vb, MR, DD, DD, 0, stream);
    launch_gemm(Qb, hourE, nullptr, qh, MR, 26, DD, 0, stream);
    launch_gemm(Qb, dayE,  nullptr, qd, MR, 8,  DD, 0, stream);
    attn_kernel<<<BB * (SS / 16), 256, 0, stream>>>(Qb, Kb, Vb, qh, qd, hourE, dayE,
                                                    in_h, in_w, seq_lens, xa);
    add_ln_kernel<<<MR, 256, 0, stream>>>(xa, x2, x1);
    launch_gemm(x1, F2a, F2ab, hf, MR, DD, DD, 1, stream);
    launch_gemm(hf, F2b, F2bb, yb, MR, DD, DD, 0, stream);
    add_ln_kernel<<<MR, 256, 0, stream>>>(yb, x1, x2);

    // ---- head ----
    rope_kernel<<<MR, 128, 0, stream>>>(x2, rx2);
    launch_gemm(user_emb, upW, upb, u, BB, DD, UU, 0, stream);
    launch_gemm(u, lhourE, nullptr, lhu, BB, 26, DD, 0, stream);
    launch_gemm(u, ldayE,  nullptr, ldu, BB, 8,  DD, 0, stream);
    rowdot_kernel<<<MR / 8, 256, 0, stream>>>(rx2, u, base);
    useragg_kernel<<<MR, 256, 0, stream>>>(rx2, base, lhu, ldu, lhourE, ldayE,
                                           in_h, in_w, lab_h, lab_w, seq_lens, aggb);
    // wide decoder GEMM: N-blocked WMMA kernel (A-fragment reuse x4)
    launch_gemm4(aggb, decW, decb, out, MR, PP, DD, stream);
}